// ObjectDetect_14259291423003
// MI455X (gfx1250) — compile-verified
//
#include <hip/hip_runtime.h>
#include <hip/hip_bf16.h>
#include <math.h>

typedef __attribute__((ext_vector_type(2))) float v2f;
typedef __attribute__((ext_vector_type(8))) float v8f;

#define NMAXK 32
#define NCK   80
#define TOPKK 10
#define EPSK  1e-9f
#define A_MAX 8400

__device__ __forceinline__ float sigmoidf_(float x) { return 1.0f / (1.0f + __expf(-x)); }

// ---------------------------------------------------------------------------
// Kernel 1: DFL decode. dist[m] = sum_k softmax(logits[m])[k] * k, k=0..15,
// computed as a 16-row x K=16 f32 matrix product against a broadcast
// projection matrix using 4 chained V_WMMA_F32_16X16X4_F32 ops.
// One wave = 4 anchors (16 rows = anchor*4+side). Lanes L and L+16 both
// compute row (L&15)'s softmax so WMMA A/B fragments are register-resident.
// blockIdx.y = batch -> no integer div/mod anywhere in the hot loop.
// ---------------------------------------------------------------------------
__global__ __launch_bounds__(256)
void k_dfl_wmma(const float* __restrict__ pred_distri,
                const float* __restrict__ anchor_points,
                const float* __restrict__ stride_tensor,
                float* __restrict__ pd_bboxes,
                int A)
{
  const int b    = blockIdx.y;
  const int lane = threadIdx.x & 31;
  const int hi   = lane >> 4;     // 0: supplies K pair {0,1} of each chunk; 1: {2,3}
  const int row  = lane & 15;     // A-matrix row fed by this lane
  const int waveId = (int)((blockIdx.x * blockDim.x + threadIdx.x) >> 5);
  const int nWaves = (int)((gridDim.x * blockDim.x) >> 5);
  const int nTiles = A >> 2;      // A divisible by 4 (A = 8400)

  const float* pdist = pred_distri + (size_t)b * A * 64;
  float*       pdout = pd_bboxes   + (size_t)b * A * 4;

  for (int tile = waveId; tile < nTiles; tile += nWaves) {
    const int base   = tile << 2;
    const int anchor = base + (row >> 2);
    const int side   = row & 3;
    const float4* p = (const float4*)(pdist + ((size_t)anchor << 6) + ((size_t)side << 4));
    float4 q0 = p[0], q1 = p[1], q2 = p[2], q3 = p[3];

    // softmax over 16 logits of this row
    float m = fmaxf(fmaxf(fmaxf(q0.x,q0.y),fmaxf(q0.z,q0.w)),
              fmaxf(fmaxf(fmaxf(q1.x,q1.y),fmaxf(q1.z,q1.w)),
              fmaxf(fmaxf(fmaxf(q2.x,q2.y),fmaxf(q2.z,q2.w)),
                    fmaxf(fmaxf(q3.x,q3.y),fmaxf(q3.z,q3.w)))));
    q0.x=__expf(q0.x-m); q0.y=__expf(q0.y-m); q0.z=__expf(q0.z-m); q0.w=__expf(q0.w-m);
    q1.x=__expf(q1.x-m); q1.y=__expf(q1.y-m); q1.z=__expf(q1.z-m); q1.w=__expf(q1.w-m);
    q2.x=__expf(q2.x-m); q2.y=__expf(q2.y-m); q2.z=__expf(q2.z-m); q2.w=__expf(q2.w-m);
    q3.x=__expf(q3.x-m); q3.y=__expf(q3.y-m); q3.z=__expf(q3.z-m); q3.w=__expf(q3.w-m);
    const float s = (q0.x+q0.y+q0.z+q0.w)+(q1.x+q1.y+q1.z+q1.w)
                  + (q2.x+q2.y+q2.z+q2.w)+(q3.x+q3.y+q3.z+q3.w);
    const float inv = 1.0f / s;
    q0.x*=inv; q0.y*=inv; q0.z*=inv; q0.w*=inv;
    q1.x*=inv; q1.y*=inv; q1.z*=inv; q1.w*=inv;
    q2.x*=inv; q2.y*=inv; q2.z*=inv; q2.w*=inv;
    q3.x*=inv; q3.y*=inv; q3.z*=inv; q3.w*=inv;

    v8f acc = {};
    // D += A_chunk(16x4) * Proj_chunk(4x16, proj value broadcast over columns)
#define DFL_CHUNK(c, q) { \
      v2f aF, bF; \
      aF[0] = hi ? (q).z : (q).x; \
      aF[1] = hi ? (q).w : (q).y; \
      const float k0 = (float)(4*(c) + 2*hi); \
      bF[0] = k0; bF[1] = k0 + 1.0f; \
      acc = __builtin_amdgcn_wmma_f32_16x16x4_f32(false, aF, false, bF, (short)0, acc, false, false); }
    DFL_CHUNK(0, q0)
    DFL_CHUNK(1, q1)
    DFL_CHUNK(2, q2)
    DFL_CHUNK(3, q3)
#undef DFL_CHUNK

    // Columns of D are identical. VGPR r: lanes 0-15 hold M=r, lanes 16-31 M=r+8.
    // Lane 0 emits anchors base+0..1, lane 16 emits base+2..3.
    if (row == 0) {
#pragma unroll
      for (int t = 0; t < 2; ++t) {
        const int aa   = base + 2*hi + t;     // anchor index within batch
        const float ax = anchor_points[2*aa+0];
        const float ay = anchor_points[2*aa+1];
        const float st = stride_tensor[aa];
        const float d0 = acc[4*t+0], d1 = acc[4*t+1], d2 = acc[4*t+2], d3 = acc[4*t+3];
        float4 o; o.x = (ax-d0)*st; o.y = (ay-d1)*st; o.z = (ax+d2)*st; o.w = (ay+d3)*st;
        *((float4*)(pdout + ((size_t)aa << 2))) = o;
      }
    }
  }
}

// ---------------------------------------------------------------------------
// Kernel 2: per (b,a) thread loops the 32 GT boxes (staged in LDS with their
// labels and precomputed arctans): one pred-box load reused 32x, per-n stores
// coalesced in a, pred-side atan hoisted out of the loop.
// ---------------------------------------------------------------------------
__global__ __launch_bounds__(256)
void k_pair(const float* __restrict__ pd_bboxes,
            const float* __restrict__ pred_scores,
            const float* __restrict__ gt_bboxes,
            const int*   __restrict__ gt_labels,
            const float* __restrict__ anchor_points,
            const float* __restrict__ stride_tensor,
            float* __restrict__ overlaps,
            float* __restrict__ amv,
            int A)
{
  __shared__ float4 s_gt[NMAXK];
  __shared__ float  s_gat[NMAXK];   // atan(w1/(h1+eps)) per gt
  __shared__ int    s_lab[NMAXK];
  const int b   = blockIdx.y;
  const int tid = threadIdx.x;
  if (tid < NMAXK) {
    const float4 g = ((const float4*)gt_bboxes)[b * NMAXK + tid];
    s_gt[tid]  = g;
    s_gat[tid] = atanf((g.z - g.x) / (g.w - g.y + 1e-7f));
    s_lab[tid] = gt_labels[b * NMAXK + tid];
  }
  __syncthreads();

  const int a = blockIdx.x * 256 + tid;
  if (a >= A) return;

  const float4 pb = ((const float4*)pd_bboxes)[(size_t)b * A + a];
  const float  st  = stride_tensor[a];
  const float  sax = anchor_points[2*a+0] * st;
  const float  say = anchor_points[2*a+1] * st;
  const float  w2 = pb.z - pb.x, h2 = pb.w - pb.y;
  const float  area2 = w2 * h2;
  const float  at2 = atanf(w2 / (h2 + 1e-7f));
  const float* srow = pred_scores + ((size_t)b * A + a) * NCK;
  float* orow = overlaps + (size_t)b * NMAXK * A + a;
  float* arow = amv      + (size_t)b * NMAXK * A + a;

  for (int n = 0; n < NMAXK; ++n) {
    const float4 g = s_gt[n];
    // CIoU(gt, pred), eps = 1e-7
    const float iw = fmaxf(fminf(g.z, pb.z) - fmaxf(g.x, pb.x), 0.0f);
    const float ih = fmaxf(fminf(g.w, pb.w) - fmaxf(g.y, pb.y), 0.0f);
    const float inter = iw * ih;
    const float w1 = g.z - g.x, h1 = g.w - g.y;
    const float uni = w1*h1 + area2 - inter + 1e-7f;
    const float iou = inter / uni;
    const float cw = fmaxf(g.z, pb.z) - fminf(g.x, pb.x);
    const float ch = fmaxf(g.w, pb.w) - fminf(g.y, pb.y);
    const float c2 = cw*cw + ch*ch + 1e-7f;
    const float dx = pb.x + pb.z - g.x - g.z;
    const float dy = pb.y + pb.w - g.y - g.w;
    const float rho2 = (dx*dx + dy*dy) * 0.25f;
    const float dat = at2 - s_gat[n];
    const float v = 0.40528473456935108577551785283891f * dat * dat; // 4/pi^2
    const float alpha = v / (v - iou + 1.0f + 1e-7f);
    const float ov = fmaxf(iou - (rho2 / c2 + v * alpha), 0.0f);

    // in_gts: scaled anchor center strictly inside gt (zeroed gt -> 0 => mask_gt folded in)
    const float dmin = fminf(fminf(sax - g.x, say - g.y), fminf(g.z - sax, g.w - say));
    const float ing  = (dmin > EPSK) ? 1.0f : 0.0f;

    const float sc = sigmoidf_(srow[s_lab[n]]);
    const float o3 = ov * ov * ov;
    const float am = sqrtf(sc) * (o3 * o3) * ing;  // score^0.5 * iou^6 * mask_cand

    orow[(size_t)n * A] = ov;
    arow[(size_t)n * A] = am;
  }
}

// ---------------------------------------------------------------------------
// Kernel 3: exact iterative top-10 per (b,n) row in LDS; ties -> lowest index
// (matches jax.lax.top_k). Invalid (value <= EPS) entries recorded as -1.
// ---------------------------------------------------------------------------
__global__ __launch_bounds__(256)
void k_topk(const float* __restrict__ amv, int* __restrict__ topk_idx, int A)
{
  __shared__ float sv[A_MAX];
  __shared__ float rv[256];
  __shared__ int   ri[256];
  const int bn  = blockIdx.x;
  const int tid = threadIdx.x;
  const float* rowp = amv + (size_t)bn * A;
  for (int j = tid; j < A; j += 256) sv[j] = rowp[j];
  __syncthreads();

  for (int it = 0; it < TOPKK; ++it) {
    float best = -1e30f; int bidx = 0x7fffffff;
    for (int j = tid; j < A; j += 256) {
      const float v = sv[j];
      if (v > best || (v == best && j < bidx)) { best = v; bidx = j; }
    }
    rv[tid] = best; ri[tid] = bidx;
    __syncthreads();
    for (int off = 128; off > 0; off >>= 1) {
      if (tid < off) {
        const float v2 = rv[tid + off]; const int i2 = ri[tid + off];
        if (v2 > rv[tid] || (v2 == rv[tid] && i2 < ri[tid])) { rv[tid] = v2; ri[tid] = i2; }
      }
      __syncthreads();
    }
    if (tid == 0) {
      const float bv = rv[0]; const int bi = ri[0];
      topk_idx[bn * TOPKK + it] = (bv > EPSK) ? bi : -1;
      if (bi >= 0 && bi < A) sv[bi] = -1e30f;
    }
    __syncthreads();
  }
}

// ---------------------------------------------------------------------------
// Kernel 4: per (b,a) build mask_pos bits, resolve multi-GT anchors by max
// overlap, and atomically accumulate per-(b,n) pos_align / pos_ov maxima
// (float-as-uint atomicMax: values are >= 0, arrays memset to 0).
// ---------------------------------------------------------------------------
__global__ __launch_bounds__(256)
void k_resolve(const float* __restrict__ overlaps,
               const float* __restrict__ amv,
               const int*   __restrict__ topk_idx,
               unsigned int* __restrict__ maskbits,
               unsigned int* __restrict__ pos_align_u,
               unsigned int* __restrict__ pos_ov_u,
               int A)
{
  __shared__ int tk[NMAXK * TOPKK];
  const int b = blockIdx.y;
  const int a = blockIdx.x * blockDim.x + threadIdx.x;
  for (int j = threadIdx.x; j < NMAXK * TOPKK; j += blockDim.x)
    tk[j] = topk_idx[b * NMAXK * TOPKK + j];
  __syncthreads();
  if (a >= A) return;

  unsigned bits = 0u;
  for (int n = 0; n < NMAXK; ++n) {
    bool hit = false;
#pragma unroll
    for (int k = 0; k < TOPKK; ++k) hit = hit || (tk[n * TOPKK + k] == a);
    if (hit) bits |= (1u << n);
  }
  const size_t rowbase = (size_t)b * NMAXK * A + a;
  if (__popc(bits) > 1) {
    float best = -1e30f; int bestn = 0;
    for (int n = 0; n < NMAXK; ++n) {
      const float v = overlaps[rowbase + (size_t)n * A];
      if (v > best) { best = v; bestn = n; }       // lowest n wins ties (argmax)
    }
    bits = 1u << bestn;
  }
  maskbits[(size_t)b * A + a] = bits;

  unsigned rem = bits;
  while (rem) {
    const int n = __ffs(rem) - 1; rem &= rem - 1;
    const float amv_ = amv[rowbase + (size_t)n * A];
    const float ovv  = overlaps[rowbase + (size_t)n * A];
    atomicMax(&pos_align_u[b * NMAXK + n], __float_as_uint(amv_));
    atomicMax(&pos_ov_u[b * NMAXK + n],    __float_as_uint(ovv));
  }
}

// ---------------------------------------------------------------------------
// Kernel 5: per (b,a) norm_align + coalesced nontemporal emission of the
// 80-class one-hot * norm row (zero rows for unassigned anchors).
// ---------------------------------------------------------------------------
__global__ __launch_bounds__(256)
void k_emit(const float* __restrict__ amv,
            const unsigned int* __restrict__ maskbits,
            const int* __restrict__ gt_labels,
            const unsigned int* __restrict__ pos_align_u,
            const unsigned int* __restrict__ pos_ov_u,
            float* __restrict__ out, int A)
{
  __shared__ float s_pa[NMAXK], s_po[NMAXK];
  __shared__ int   s_lab[NMAXK];
  __shared__ float s_norm[256];
  __shared__ int   s_tlab[256];
  const int b = blockIdx.y;
  const int tid = threadIdx.x;
  if (tid < NMAXK) {
    s_pa[tid]  = __uint_as_float(pos_align_u[b * NMAXK + tid]);
    s_po[tid]  = __uint_as_float(pos_ov_u[b * NMAXK + tid]);
    s_lab[tid] = gt_labels[b * NMAXK + tid];
  }
  __syncthreads();

  const int a0 = blockIdx.x * 256;
  const int a  = a0 + tid;
  float norm = 0.0f; int tl = 0;
  if (a < A) {
    unsigned bits = maskbits[(size_t)b * A + a];
    if (bits) {
      tl = s_lab[__ffs(bits) - 1];                 // argmax(mask_pos) = first set n
      const size_t rowbase = (size_t)b * NMAXK * A + a;
      unsigned rem = bits;
      while (rem) {
        const int n = __ffs(rem) - 1; rem &= rem - 1;
        const float val = amv[rowbase + (size_t)n * A] * s_po[n] / (s_pa[n] + EPSK);
        norm = fmaxf(norm, val);
      }
    }
  }
  s_norm[tid] = norm; s_tlab[tid] = tl;
  __syncthreads();

  const int count = min(256, A - a0);
  float* obase = out + ((size_t)b * A + a0) * NCK;
  const int total = count * NCK;
  for (int e = tid; e < total; e += 256) {
    const int la = e / NCK;
    const int c  = e - la * NCK;
    const float w = (c == s_tlab[la]) ? s_norm[la] : 0.0f;
    __builtin_nontemporal_store(w, obase + e);
  }
}

// ---------------------------------------------------------------------------
extern "C" void kernel_launch(void* const* d_in, const int* in_sizes, int n_in,
                              void* d_out, int out_size, void* d_ws, size_t ws_size,
                              hipStream_t stream)
{
  (void)n_in; (void)out_size; (void)ws_size;
  const float* pred_distri   = (const float*)d_in[0];
  const float* pred_scores   = (const float*)d_in[1];
  const float* gt_bboxes     = (const float*)d_in[2];
  const float* anchor_points = (const float*)d_in[3];
  const float* stride_tensor = (const float*)d_in[4];
  const int*   gt_labels     = (const int*)d_in[5];
  // d_in[6] = mask_gt: redundant (masked gt boxes are all-zero => in_gts == 0)

  const int A  = in_sizes[3] / 2;
  const int B  = in_sizes[2] / (NMAXK * 4);
  const int BA = B * A;

  char* ws = (char*)d_ws;
  float*        pd = (float*)ws;        ws += (size_t)BA * 4 * sizeof(float);
  float*        ov = (float*)ws;        ws += (size_t)B * NMAXK * A * sizeof(float);
  float*        am = (float*)ws;        ws += (size_t)B * NMAXK * A * sizeof(float);
  int*          tk = (int*)ws;          ws += (size_t)B * NMAXK * TOPKK * sizeof(int);
  unsigned int* mb = (unsigned int*)ws; ws += (size_t)BA * sizeof(unsigned int);
  unsigned int* pa = (unsigned int*)ws; ws += (size_t)B * NMAXK * sizeof(unsigned int);
  unsigned int* po = (unsigned int*)ws; ws += (size_t)B * NMAXK * sizeof(unsigned int);

  hipMemsetAsync(pa, 0, 2 * (size_t)B * NMAXK * sizeof(unsigned int), stream); // pa+po contiguous

  dim3 g1(128, B);
  k_dfl_wmma<<<g1, 256, 0, stream>>>(pred_distri, anchor_points, stride_tensor, pd, A);

  dim3 gAB((A + 255) / 256, B);
  k_pair<<<gAB, 256, 0, stream>>>(pd, pred_scores, gt_bboxes, gt_labels,
                                  anchor_points, stride_tensor, ov, am, A);

  k_topk<<<B * NMAXK, 256, 0, stream>>>(am, tk, A);

  k_resolve<<<gAB, 256, 0, stream>>>(ov, am, tk, mb, pa, po, A);
  k_emit<<<gAB, 256, 0, stream>>>(am, mb, gt_labels, pa, po, (float*)d_out, A);
}